// CosineSimilarity_11587821764946
// MI455X (gfx1250) — compile-verified
//
#include <hip/hip_runtime.h>
#include <hip/hip_bf16.h>

typedef float v2f __attribute__((ext_vector_type(2)));
typedef float v8f __attribute__((ext_vector_type(8)));

#define DDIM 1024
#define ROWS_PER_WAVE 16
#define WAVES_PER_BLOCK 8
#define ROWS_PER_BLOCK (ROWS_PER_WAVE * WAVES_PER_BLOCK)  // 128

// ---------------------------------------------------------------------------
// Kernel 1: q_hat = q / max(||q||, eps).  One block of 256 threads (8 waves).
// ---------------------------------------------------------------------------
__global__ void cosine_qnorm_kernel(const float* __restrict__ q,
                                    float* __restrict__ qn) {
    __shared__ float red[WAVES_PER_BLOCK];
    const int t = threadIdx.x;                 // 0..255
    float4 v = ((const float4*)q)[t];          // 256*4 = 1024 = DDIM
    float ss = v.x * v.x + v.y * v.y + v.z * v.z + v.w * v.w;
    // wave32 reduction
    #pragma unroll
    for (int o = 16; o > 0; o >>= 1)
        ss += __shfl_xor(ss, o, 32);
    if ((t & 31) == 0) red[t >> 5] = ss;
    __syncthreads();
    float tot = 0.0f;
    #pragma unroll
    for (int w = 0; w < WAVES_PER_BLOCK; ++w) tot += red[w];
    const float inv = 1.0f / fmaxf(sqrtf(tot), 1e-12f);
    float4 o4;
    o4.x = v.x * inv; o4.y = v.y * inv; o4.z = v.z * inv; o4.w = v.w * inv;
    ((float4*)qn)[t] = o4;
}

// ---------------------------------------------------------------------------
// Kernel 2: fused dot(W[i], q_hat) and ||W[i]|| in one pass over W.
// Each wave handles 16 rows via V_WMMA_F32_16X16X4_F32 (fp32-accurate).
//
// K-permutation trick for max-width loads: per 8-column group, lane-lo
// loads W cols [k..k+3] (b128), lane-hi loads [k+4..k+7]. Since a dot
// product is invariant under a column permutation applied to BOTH A and B,
//   WMMA step 0 uses .xy of both halves -> true cols {k,k+1,k+4,k+5}
//   WMMA step 1 uses .zw               -> true cols {k+2,k+3,k+6,k+7}
// with the B fragment (q_hat from LDS) using the identical +4*hi mapping.
// ---------------------------------------------------------------------------
__global__ void __launch_bounds__(256)
cosine_sim_kernel(const float* __restrict__ W,
                  const float* __restrict__ qn_g,
                  float* __restrict__ out, int N) {
    __shared__ float qs[DDIM];
    __shared__ float dots[WAVES_PER_BLOCK][ROWS_PER_WAVE];

    const int t = threadIdx.x;
    // Stage q_hat into LDS (4 KB), coalesced float4.
    ((float4*)qs)[t] = ((const float4*)qn_g)[t];
    __syncthreads();

    const int wave = t >> 5;        // 0..7
    const int lane = t & 31;        // 0..31
    const int m    = lane & 15;     // row within tile this lane loads for A
    const int hi   = lane >> 4;     // 0: cols k..k+3, 1: cols k+4..k+7

    const int row0 = (int)blockIdx.x * ROWS_PER_BLOCK + wave * ROWS_PER_WAVE;
    int rowLoad = row0 + m;
    if (rowLoad > N - 1) rowLoad = N - 1;     // branchless clamp (EXEC all-1s)

    const float* __restrict__ wrow = W + (size_t)rowLoad * DDIM + 4 * hi;
    const float* __restrict__ qrow = qs + 4 * hi;

    v8f c = {0.f, 0.f, 0.f, 0.f, 0.f, 0.f, 0.f, 0.f};
    float ss = 0.0f;

    #pragma unroll 4
    for (int k = 0; k < DDIM; k += 8) {
        float4 a4 = *(const float4*)(wrow + k);   // global_load_b128, row stream
        float4 b4 = *(const float4*)(qrow + k);   // ds_load_b128, half-broadcast
        v2f a0 = {a4.x, a4.y}, a1 = {a4.z, a4.w};
        v2f b0 = {b4.x, b4.y}, b1 = {b4.z, b4.w};
        c = __builtin_amdgcn_wmma_f32_16x16x4_f32(
                false, a0, false, b0, (short)0, c, false, false);
        c = __builtin_amdgcn_wmma_f32_16x16x4_f32(
                false, a1, false, b1, (short)0, c, false, false);
        ss = fmaf(a4.x, a4.x, ss);
        ss = fmaf(a4.y, a4.y, ss);
        ss = fmaf(a4.z, a4.z, ss);
        ss = fmaf(a4.w, a4.w, ss);
    }

    // Combine the two column-halves of the row-norm: lanes L and L^16 share
    // row m (lane-lo covered cols 0..3 mod 8, lane-hi cols 4..7 mod 8).
    ss += __shfl_xor(ss, 16, 32);

    // Every column of D equals dot(row M, q_hat). Lane 0 holds M=0..7 in its
    // 8 accumulator VGPRs, lane 16 holds M=8..15. Spill via LDS.
    if ((lane & 15) == 0) {
        #pragma unroll
        for (int v = 0; v < 8; ++v)
            dots[wave][hi * 8 + v] = c[v];
    }
    __syncthreads();

    if (lane < 16) {
        const int r = row0 + lane;
        if (r < N) {
            const float dot  = dots[wave][lane];
            const float norm = fmaxf(sqrtf(ss), 1e-12f);  // ss is row 'lane'
            out[r] = dot / norm;
        }
    }
}

// ---------------------------------------------------------------------------
extern "C" void kernel_launch(void* const* d_in, const int* in_sizes, int n_in,
                              void* d_out, int out_size, void* d_ws, size_t ws_size,
                              hipStream_t stream) {
    const float* q = (const float*)d_in[0];   // [D] fp32
    const float* W = (const float*)d_in[1];   // [N, D] fp32
    float* out = (float*)d_out;               // [N] fp32
    float* qn  = (float*)d_ws;                // workspace: normalized query [D]
    const int N = out_size;

    cosine_qnorm_kernel<<<dim3(1), dim3(256), 0, stream>>>(q, qn);

    const int blocks = (N + ROWS_PER_BLOCK - 1) / ROWS_PER_BLOCK;
    cosine_sim_kernel<<<dim3(blocks), dim3(256), 0, stream>>>(W, qn, out, N);
}